// Net_autoencpsdhigh_52467320488355
// MI455X (gfx1250) — compile-verified
//
#include <hip/hip_runtime.h>
#include <math.h>

typedef float v2f __attribute__((ext_vector_type(2)));
typedef float v8f __attribute__((ext_vector_type(8)));

#define B_    64
#define P_    12273
#define J_    80
#define MOT_  94
#define KF_   64
#define H_    128
#define NBWN  340
#define J4    320
#define MOTK  6016           // MOT*K
#define C3    36819          // P*3
#define DPSDN (NBWN * C3)    // 12,518,460

// workspace layout (floats)
#define WS_L1    0
#define WS_SQ    1
#define WS_DKEYT 8                       // pair-interleaved [170][64][2]
#define WS_GT    (WS_DKEYT + NBWN * B_)  // pair-interleaved [3][160][64][2]

__device__ __forceinline__ v8f wmma4(v2f a, v2f b, v8f c) {
  return __builtin_amdgcn_wmma_f32_16x16x4_f32(false, a, false, b, (short)0, c,
                                               false, false);
}

// ---------------- k0: zero loss accumulators ----------------
__global__ void k0_init(float* __restrict__ ws) {
  if (threadIdx.x == 0) { ws[WS_L1] = 0.f; ws[WS_SQ] = 0.f; }
}

// ---------------- k1: field_input + MLP + euler->R, build GT ----------------
// GT pair-interleaved: GT[x*J4*B_ + (q>>1)*2*B_ + b*2 + (q&1)], q = jj*4+y
__global__ void k1_pose(const float* __restrict__ query,
                        const float* __restrict__ mw,
                        const float* __restrict__ tmtemp,
                        const float* __restrict__ W1, const float* __restrict__ b1,
                        const float* __restrict__ W2, const float* __restrict__ b2,
                        const float* __restrict__ pose_std,
                        const float* __restrict__ pose_mean,
                        const float* __restrict__ trans_std,
                        const float* __restrict__ trans_mean,
                        float* __restrict__ GT) {
  __shared__ float h[3 + KF_];
  __shared__ float hid[H_];
  __shared__ float pr[6];
  const int jj = blockIdx.x, b = blockIdx.y;
  const int t = threadIdx.x;

  if (t < KF_) {  // field_input[b, jj, k]
    float acc = 0.f;
    const float* tm = tmtemp + b * MOT_ * KF_ + t;
    const float* w = mw + jj * MOT_;
    for (int m = 0; m < MOT_; ++m) {
      float wv = w[m];
      wv = wv > 0.f ? wv : 0.f;
      acc += tm[m * KF_] * wv;
    }
    h[3 + t] = acc;
  } else if (t < KF_ + 3) {
    h[t - KF_] = query[(b * J_ + jj) * 3 + (t - KF_)];
  }
  __syncthreads();
  {  // hidden = relu(h @ W1 + b1), all 128 threads
    float acc = b1[t];
    for (int i = 0; i < 3 + KF_; ++i) acc += h[i] * W1[i * H_ + t];
    hid[t] = acc > 0.f ? acc : 0.f;
  }
  __syncthreads();
  if (t < 6) {  // pred_rt = hid @ W2 + b2
    float acc = b2[t];
    for (int o = 0; o < H_; ++o) acc += hid[o] * W2[o * 6 + t];
    pr[t] = acc;
  }
  __syncthreads();
  if (t == 0) {
    const float d2r = 0.017453292519943295f;
    float ax = (pr[0] * pose_std[0] + pose_mean[0]) * d2r;
    float ay = (pr[1] * pose_std[1] + pose_mean[1]) * d2r;
    float az = (pr[2] * pose_std[2] + pose_mean[2]) * d2r;
    float tr[3];
    for (int x = 0; x < 3; ++x)
      tr[x] = (h[x] + pr[3 + x]) * trans_std[x] + trans_mean[x];
    float sx = sinf(ax), cx = cosf(ax);
    float sy = sinf(ay), cy = cosf(ay);
    float sz = sinf(az), cz = cosf(az);
    float M[3][4];
    M[0][0] = cz * cy; M[0][1] = cz * sy * sx - sz * cx; M[0][2] = cz * sy * cx + sz * sx;
    M[1][0] = sz * cy; M[1][1] = sz * sy * sx + cz * cx; M[1][2] = sz * sy * cx - cz * sx;
    M[2][0] = -sy;     M[2][1] = cy * sx;                M[2][2] = cy * cx;
    M[0][3] = tr[0]; M[1][3] = tr[1]; M[2][3] = tr[2];
    for (int x = 0; x < 3; ++x)
      for (int y = 0; y < 4; ++y) {
        const int q = jj * 4 + y;
        GT[x * J4 * B_ + (q >> 1) * 2 * B_ + b * 2 + (q & 1)] = M[x][y];
      }
  }
}

// ---------------- k2: dkeyT = (tmtemp_flat @ Wd + bd)^T via WMMA ------------
// dkeyT pair-interleaved: dkeyT[(n>>1)*2*B_ + b*2 + (n&1)]
__global__ void k2_dkey(const float* __restrict__ tmtemp,
                        const float* __restrict__ Wd,
                        const float* __restrict__ bd,
                        float* __restrict__ dkeyT) {
  const int t = threadIdx.x;
  const int wave = t >> 5, lane = t & 31;
  const int row = lane & 15, hi = lane >> 4;
  const int n = blockIdx.x * 16 + row;
  const int nc = n < NBWN ? n : (NBWN - 1);  // clamp: bad cols never stored
  v8f acc = {};
  const float* ta = tmtemp + (wave * 16 + row) * MOTK;
#pragma unroll 4
  for (int s = 0; s < MOTK / 4; ++s) {
    const int k0 = 4 * s + 2 * hi;
    v2f a, bv;
    a.x = ta[k0];
    a.y = ta[k0 + 1];
    bv.x = Wd[k0 * NBWN + nc];
    bv.y = Wd[(k0 + 1) * NBWN + nc];
    acc = wmma4(a, bv, acc);
  }
  if (n < NBWN) {
    const float bias = bd[n];
    for (int i = 0; i < 8; ++i) {
      const int br = wave * 16 + hi * 8 + i;
      dkeyT[(n >> 1) * 2 * B_ + br * 2 + (n & 1)] = acc[i] + bias;
    }
  }
}

// ---------------- k3: fused detail GEMM + skin GEMM + out_pc + losses -------
__global__ void k3_main(const float* __restrict__ in_pc,
                        const float* __restrict__ rest_verts,
                        const float* __restrict__ sw,
                        const float* __restrict__ DPSD,
                        const float* __restrict__ res_std,
                        const float* __restrict__ res_mean,
                        const float* __restrict__ dkeyT,
                        const float* __restrict__ GT,
                        float* __restrict__ out,
                        float* __restrict__ ws) {
  __shared__ float f_lds[J4 * 16];  // pair-interleaved [(q>>1)][n][2]
  __shared__ float red[128];
  const int t = threadIdx.x;
  const int wave = t >> 5, lane = t & 31;
  const int row = lane & 15, hi = lane >> 4;
  const int p0 = blockIdx.x * 16;
  const int x = blockIdx.y;

  // build f tile: f[q = j*4+y][p0+n] = sw[p,j] * [v_p, 1][y]
  for (int idx = t; idx < J4 * 16; idx += 128) {
    const int n = idx & 15, q = idx >> 4;
    const int j = q >> 2, y = q & 3;
    const int p = p0 + n;
    float v = 0.f;
    if (p < P_) {
      const float e = (y < 3) ? rest_verts[p * 3 + y] : 1.0f;
      v = sw[p * J_ + j] * e;
    }
    f_lds[(q >> 1) * 32 + n * 2 + (q & 1)] = v;
  }
  __syncthreads();

  const float std_x = res_std[x], mean_x = res_mean[x];
  const int p = p0 + row;
  const bool pv = p < P_;
  // clamp column: invalid columns only affect unstored outputs
  const int coln = (pv ? p : (P_ - 1)) * 3 + x;
  const float sqmsk = (pv && wave == 0) ? 1.f : 0.f;
  const int m = wave * 16 + row;

  const v2f* dk2 = (const v2f*)dkeyT;
  const v2f* g2 = (const v2f*)(GT + x * J4 * B_);
  const v2f* f2 = (const v2f*)f_lds;

  // detail: acc_d[b, p] = sum_k dkey[b,k] * DPSD[k, p*3+x]
  v8f accd = {};
  float sq = 0.f;  // wave 0 touches each DPSD element exactly once
#pragma unroll 5
  for (int s = 0; s < NBWN / 4; ++s) {
    const int k0 = 4 * s + 2 * hi;
    v2f bv;
    const v2f a = dk2[(k0 >> 1) * B_ + m];
    bv.x = DPSD[k0 * C3 + coln];
    bv.y = DPSD[(k0 + 1) * C3 + coln];
    sq += (bv.x * bv.x + bv.y * bv.y) * sqmsk;
    accd = wmma4(a, bv, accd);
  }
  // skinning: acc_s[b, p] = sum_q G[b, q, x] * f[q, p]
  v8f accs = {};
#pragma unroll 5
  for (int s = 0; s < J4 / 4; ++s) {
    const int k0 = 4 * s + 2 * hi;
    const v2f a = g2[(k0 >> 1) * B_ + m];
    const v2f bv = f2[(k0 >> 1) * 16 + row];
    accs = wmma4(a, bv, accs);
  }

  float lsum = 0.f;
  if (pv) {
    for (int i = 0; i < 8; ++i) {
      const int br = wave * 16 + hi * 8 + i;
      const float val = accs[i] + accd[i] * std_x + mean_x;
      const int oidx = (br * P_ + p) * 3 + x;
      out[oidx] = val;
      lsum += fabsf(in_pc[oidx] - val);
    }
  }
  red[t] = lsum;
  __syncthreads();
  for (int off = 64; off > 0; off >>= 1) {
    if (t < off) red[t] += red[t + off];
    __syncthreads();
  }
  if (t == 0) atomicAdd(&ws[WS_L1], red[0]);
  __syncthreads();
  red[t] = sq;
  __syncthreads();
  for (int off = 64; off > 0; off >>= 1) {
    if (t < off) red[t] += red[t + off];
    __syncthreads();
  }
  if (t == 0) atomicAdd(&ws[WS_SQ], red[0]);
}

// ---------------- k5: finalize loss ----------------
__global__ void k5_fin(const float* __restrict__ ws, float* __restrict__ out) {
  if (threadIdx.x == 0)
    out[0] = ws[WS_L1] * (1.0f / (float)(B_ * P_ * 3)) +
             1e-4f * ws[WS_SQ] * (1.0f / (float)DPSDN);
}

extern "C" void kernel_launch(void* const* d_in, const int* in_sizes, int n_in,
                              void* d_out, int out_size, void* d_ws, size_t ws_size,
                              hipStream_t stream) {
  const float* in_pc      = (const float*)d_in[0];
  const float* rest_verts = (const float*)d_in[2];
  const float* skin_w     = (const float*)d_in[3];
  const float* mw         = (const float*)d_in[4];
  const float* query      = (const float*)d_in[5];
  const float* pose_std   = (const float*)d_in[6];
  const float* pose_mean  = (const float*)d_in[7];
  const float* trans_std  = (const float*)d_in[8];
  const float* trans_mean = (const float*)d_in[9];
  const float* W1         = (const float*)d_in[10];
  const float* b1         = (const float*)d_in[11];
  const float* W2         = (const float*)d_in[12];
  const float* b2         = (const float*)d_in[13];
  const float* tmtemp     = (const float*)d_in[14];
  const float* Wd         = (const float*)d_in[15];
  const float* bd         = (const float*)d_in[16];
  const float* DPSD       = (const float*)d_in[17];
  const float* res_std    = (const float*)d_in[18];
  const float* res_mean   = (const float*)d_in[19];
  float* ws  = (float*)d_ws;
  float* out = (float*)d_out;

  k0_init<<<1, 64, 0, stream>>>(ws);
  k1_pose<<<dim3(J_, B_), 128, 0, stream>>>(query, mw, tmtemp, W1, b1, W2, b2,
                                            pose_std, pose_mean, trans_std,
                                            trans_mean, ws + WS_GT);
  k2_dkey<<<(NBWN + 15) / 16, 128, 0, stream>>>(tmtemp, Wd, bd, ws + WS_DKEYT);
  k3_main<<<dim3((P_ + 15) / 16, 3), 128, 0, stream>>>(
      in_pc, rest_verts, skin_w, DPSD, res_std, res_mean, ws + WS_DKEYT,
      ws + WS_GT, out + 1, ws);
  k5_fin<<<1, 32, 0, stream>>>(ws, out);
}